// SAGEConvLayer_21663815041135
// MI455X (gfx1250) — compile-verified
//
#include <hip/hip_runtime.h>

#define NN 100000
#define NE 1600000
#define DIM 128
#define BN_EPS 1e-5f

typedef __attribute__((ext_vector_type(2))) float v2f;
typedef __attribute__((ext_vector_type(8))) float v8f;

// ---------------------------------------------------------------------------
// Kernel 1: zero the workspace region (agg + deg + gsum + gsq + coefA + coefB)
// ---------------------------------------------------------------------------
__global__ void zero_kernel(float* __restrict__ p, size_t n) {
    size_t i = ((size_t)blockIdx.x * blockDim.x + threadIdx.x) * 4;
    if (i + 3 < n) {
        float4 z; z.x = 0.f; z.y = 0.f; z.z = 0.f; z.w = 0.f;
        *(float4*)(p + i) = z;
    } else {
        for (size_t j = i; j < n; ++j) p[j] = 0.f;
    }
}

// ---------------------------------------------------------------------------
// Kernel 2: one wave (32 lanes) per edge. Each lane moves a float4 of the
// 128-float source row -> coalesced 512B gather, coalesced atomic scatter.
// ---------------------------------------------------------------------------
__global__ __launch_bounds__(256) void scatter_kernel(
    const float* __restrict__ x, const int* __restrict__ ei,
    float* __restrict__ agg, float* __restrict__ deg) {
    int wid  = blockIdx.x * 8 + (threadIdx.x >> 5);   // edge id
    int lane = threadIdx.x & 31;
    if (wid >= NE) return;
    int src = ei[wid];        // edge_index[0][e]
    int dst = ei[NE + wid];   // edge_index[1][e]
    const float4 v = *(const float4*)(x + (size_t)src * DIM + lane * 4);
    float* dp = agg + (size_t)dst * DIM + lane * 4;
    atomicAdd(dp + 0, v.x);
    atomicAdd(dp + 1, v.y);
    atomicAdd(dp + 2, v.z);
    atomicAdd(dp + 3, v.w);
    if (lane == 0) atomicAdd(deg + dst, 1.0f);
}

// ---------------------------------------------------------------------------
// Kernel 3: mean = agg / max(deg,1)   (in place on agg)
// ---------------------------------------------------------------------------
__global__ void mean_kernel(float* __restrict__ agg, const float* __restrict__ deg) {
    size_t i = (size_t)blockIdx.x * blockDim.x + threadIdx.x;   // float4 index
    if (i >= (size_t)NN * DIM / 4) return;
    int node = (int)((i * 4) >> 7);                             // /DIM
    float inv = 1.0f / fmaxf(deg[node], 1.0f);
    float4 v = ((float4*)agg)[i];
    v.x *= inv; v.y *= inv; v.z *= inv; v.w *= inv;
    ((float4*)agg)[i] = v;
}

// ---------------------------------------------------------------------------
// Kernel 4: h = relu(mean@Wl + bl + x@Wr) via V_WMMA_F32_16X16X4_F32,
// weights staged in LDS (129KB of the 320KB/WGP), fused BN partial sums.
// 8 waves/block, one 16-row tile per wave, 8 N-tiles of 16 cols each.
// ---------------------------------------------------------------------------
__global__ __launch_bounds__(256) void gemm_bn_kernel(
    const float* __restrict__ mean, const float* __restrict__ x,
    const float* __restrict__ Wl, const float* __restrict__ bl,
    const float* __restrict__ Wr, float* __restrict__ h,
    float* __restrict__ gsum, float* __restrict__ gsq) {
    __shared__ float lWl[DIM * DIM];   // 64 KB
    __shared__ float lWr[DIM * DIM];   // 64 KB
    __shared__ float lsum[DIM];
    __shared__ float lsq[DIM];

    const int tid = threadIdx.x;
    for (int i = tid * 4; i < DIM * DIM; i += 256 * 4) {
        *(float4*)(lWl + i) = *(const float4*)(Wl + i);
        *(float4*)(lWr + i) = *(const float4*)(Wr + i);
    }
    if (tid < DIM) { lsum[tid] = 0.f; lsq[tid] = 0.f; }
    __syncthreads();

    const int wave = tid >> 5;
    const int lane = tid & 31;
    const int tile = blockIdx.x * 8 + wave;          // 16-row output tile
    const bool active = (tile * 16) < NN;            // wave-uniform

    if (active) {
        const int row0 = tile * 16;
        const int m  = lane & 15;                    // A row within tile
        const int kh = lane >> 4;                    // K-half select
        const float* meanRow = mean + (size_t)(row0 + m) * DIM;
        const float* xRow    = x    + (size_t)(row0 + m) * DIM;

        for (int nt = 0; nt < 8; ++nt) {
            const int ncol = nt * 16 + (lane & 15);
            const float bias = bl[ncol];
            v8f c;
#pragma unroll
            for (int r = 0; r < 8; ++r) c[r] = bias;

            // mean @ Wl
#pragma unroll
            for (int k0 = 0; k0 < DIM; k0 += 4) {
                const int ka = k0 + 2 * kh;
                v2f a; a.x = meanRow[ka];           a.y = meanRow[ka + 1];
                v2f b; b.x = lWl[ka * DIM + ncol];  b.y = lWl[(ka + 1) * DIM + ncol];
                c = __builtin_amdgcn_wmma_f32_16x16x4_f32(
                        false, a, false, b, (short)0, c, false, false);
            }
            // x @ Wr (accumulate into same C)
#pragma unroll
            for (int k0 = 0; k0 < DIM; k0 += 4) {
                const int ka = k0 + 2 * kh;
                v2f a; a.x = xRow[ka];              a.y = xRow[ka + 1];
                v2f b; b.x = lWr[ka * DIM + ncol];  b.y = lWr[(ka + 1) * DIM + ncol];
                c = __builtin_amdgcn_wmma_f32_16x16x4_f32(
                        false, a, false, b, (short)0, c, false, false);
            }

            // ReLU + store h + per-column partial sums
            float s = 0.f, s2 = 0.f;
#pragma unroll
            for (int r = 0; r < 8; ++r) {
                float v = c[r] > 0.f ? c[r] : 0.f;
                const int row = row0 + r + 8 * kh;   // D layout: VGPR r, lanes16-31 -> M=r+8
                h[(size_t)row * DIM + ncol] = v;
                s  += v;
                s2 += v * v;
            }
            // lanes l and l+16 hold the same column: fold halves, then LDS atomic
            s  += __shfl_xor(s, 16, 32);
            s2 += __shfl_xor(s2, 16, 32);
            if (lane < 16) {
                atomicAdd(&lsum[ncol], s);
                atomicAdd(&lsq[ncol], s2);
            }
        }
    }
    __syncthreads();
    if (tid < DIM) {
        atomicAdd(gsum + tid, lsum[tid]);
        atomicAdd(gsq  + tid, lsq[tid]);
    }
}

// ---------------------------------------------------------------------------
// Kernel 5: fold BN stats into per-column affine coefficients
// ---------------------------------------------------------------------------
__global__ void stats_kernel(const float* __restrict__ gsum, const float* __restrict__ gsq,
                             const float* __restrict__ gamma, const float* __restrict__ beta,
                             float* __restrict__ cA, float* __restrict__ cB) {
    int c = threadIdx.x;
    if (c >= DIM) return;
    const float invN = 1.0f / (float)NN;
    float mu  = gsum[c] * invN;
    float var = gsq[c] * invN - mu * mu;          // biased var == E[h^2]-mu^2
    float a   = rsqrtf(var + BN_EPS) * gamma[c];
    cA[c] = a;
    cB[c] = beta[c] - mu * a;
}

// ---------------------------------------------------------------------------
// Kernel 6: out = x + h*a[col] + b[col]   (h lives in d_out, overwritten in place)
// ---------------------------------------------------------------------------
__global__ void apply_kernel(const float* __restrict__ x,
                             const float* __restrict__ cA, const float* __restrict__ cB,
                             float* __restrict__ out) {
    size_t i = (size_t)blockIdx.x * blockDim.x + threadIdx.x;   // float4 index
    if (i >= (size_t)NN * DIM / 4) return;
    int col4 = (int)(i & 31) * 4;                               // (i*4) % 128
    float4 hv = ((float4*)out)[i];
    float4 xv = ((const float4*)x)[i];
    float4 a  = *(const float4*)(cA + col4);
    float4 b  = *(const float4*)(cB + col4);
    float4 o;
    o.x = xv.x + hv.x * a.x + b.x;
    o.y = xv.y + hv.y * a.y + b.y;
    o.z = xv.z + hv.z * a.z + b.z;
    o.w = xv.w + hv.w * a.w + b.w;
    ((float4*)out)[i] = o;
}

// ---------------------------------------------------------------------------
extern "C" void kernel_launch(void* const* d_in, const int* in_sizes, int n_in,
                              void* d_out, int out_size, void* d_ws, size_t ws_size,
                              hipStream_t stream) {
    const float* x     = (const float*)d_in[0];
    const int*   ei    = (const int*)d_in[1];   // edge_index flattened [2, NE]
    const float* Wl    = (const float*)d_in[2];
    const float* bl    = (const float*)d_in[3];
    const float* Wr    = (const float*)d_in[4];
    const float* gamma = (const float*)d_in[5];
    const float* beta  = (const float*)d_in[6];
    float* out = (float*)d_out;

    // workspace layout
    float* agg  = (float*)d_ws;                 // [NN, DIM] agg -> mean (in place)
    float* deg  = agg + (size_t)NN * DIM;       // [NN]
    float* gsum = deg + NN;                     // [DIM]
    float* gsq  = gsum + DIM;                   // [DIM]
    float* cA   = gsq + DIM;                    // [DIM]
    float* cB   = cA + DIM;                     // [DIM]

    const size_t zero_elems = (size_t)NN * DIM + NN + 4 * DIM;
    int zb = (int)((zero_elems / 4 + 255) / 256);
    zero_kernel<<<zb, 256, 0, stream>>>(agg, zero_elems);

    scatter_kernel<<<NE / 8, 256, 0, stream>>>(x, ei, agg, deg);

    int mb = (int)(((size_t)NN * DIM / 4 + 255) / 256);
    mean_kernel<<<mb, 256, 0, stream>>>(agg, deg);

    const int tiles = NN / 16;                  // 6250, exact
    gemm_bn_kernel<<<(tiles + 7) / 8, 256, 0, stream>>>(agg, x, Wl, bl, Wr,
                                                        out, gsum, gsq);

    stats_kernel<<<1, 128, 0, stream>>>(gsum, gsq, gamma, beta, cA, cB);

    apply_kernel<<<mb, 256, 0, stream>>>(x, cA, cB, out);
}